// GCNDecoder_67035849556597
// MI455X (gfx1250) — compile-verified
//
#include <hip/hip_runtime.h>

typedef float v2f __attribute__((ext_vector_type(2)));
typedef float v8f __attribute__((ext_vector_type(8)));

#define HID 64

// ---------- utility: fill a float buffer with a constant ----------
__global__ void k_fill(float* __restrict__ p, long long n, float v) {
    long long i = (long long)blockIdx.x * blockDim.x + threadIdx.x;
    if (i < n) p[i] = v;
}

// ---------- degree count: deg[dst] += 1 per edge ----------
__global__ void k_count_deg(const int* __restrict__ dst, float* __restrict__ deg, int E) {
    int e = blockIdx.x * blockDim.x + threadIdx.x;
    if (e < E) atomicAdd(&deg[dst[e]], 1.0f);
}

// ---------- deg -> rsqrt(deg) in place ----------
__global__ void k_rsqrt(float* __restrict__ d, int n) {
    int i = blockIdx.x * blockDim.x + threadIdx.x;
    if (i < n) d[i] = rsqrtf(d[i]);
}

// ---------- C[M x 64] = A[M x K] @ W[K x 64] via v_wmma_f32_16x16x4_f32 ----------
// Block = 128 threads = 4 waves computing a 64-row x 64-col output macro-tile.
// Wave w owns columns [16w, 16w+16) and keeps 4 accumulators (M sub-tiles),
// so each B fragment (2 global loads) feeds 4 WMMAs -> 4x less W traffic.
// A panel (64 rows x 64 K) staged in LDS (16 KB) with coalesced chunk loads.
// Fragment layouts per CDNA5 ISA 7.12.2:
//   A 16x4 : lanes 0-15 -> K={0,1} in v0/v1 ; lanes 16-31 -> K={2,3}
//   B 4x16 : v0 = K-row (0|2), v1 = K-row (1|3), N = lane&15
//   C 16x16: VGPR r -> M=r (lanes 0-15) / M=r+8 (lanes 16-31)
__global__ __launch_bounds__(128)
void k_gemm64(const float* __restrict__ A, const float* __restrict__ W,
              float* __restrict__ C, int M, int K) {
    __shared__ float lds[64 * 64];            // 16 KB
    const int m0   = blockIdx.x * 64;
    const int lane = threadIdx.x & 31;
    const int wave = threadIdx.x >> 5;        // N tile id (0..3)
    const int ncol = (wave << 4) + (lane & 15);
    const int krow = (lane >> 4) << 1;        // 0 for lanes 0-15, 2 for lanes 16-31
    const int arow = lane & 15;

    v8f c0 = {0.f,0.f,0.f,0.f,0.f,0.f,0.f,0.f};
    v8f c1 = c0, c2 = c0, c3 = c0;

    for (int kc = 0; kc < K; kc += 64) {
        // cooperative coalesced load of A[m0..m0+63][kc..kc+63]
        #pragma unroll
        for (int i = 0; i < 32; ++i) {
            int idx = threadIdx.x + (i << 7);        // 0..4095
            int r = idx >> 6, col = idx & 63;
            int gr = m0 + r; if (gr >= M) gr = M - 1;  // clamp tail rows
            lds[idx] = A[(size_t)gr * K + kc + col];
        }
        __syncthreads();
        #pragma unroll
        for (int k4 = 0; k4 < 64; k4 += 4) {
            v2f b;
            const float* wp = W + (size_t)(kc + k4 + krow) * HID + ncol;
            b.x = wp[0];
            b.y = wp[HID];
            const float* lp = &lds[arow * 64 + k4 + krow];
            v2f a0, a1, a2, a3;
            a0.x = lp[0];          a0.y = lp[1];
            a1.x = lp[16 * 64];    a1.y = lp[16 * 64 + 1];
            a2.x = lp[32 * 64];    a2.y = lp[32 * 64 + 1];
            a3.x = lp[48 * 64];    a3.y = lp[48 * 64 + 1];
            c0 = __builtin_amdgcn_wmma_f32_16x16x4_f32(false, a0, false, b, (short)0, c0, false, false);
            c1 = __builtin_amdgcn_wmma_f32_16x16x4_f32(false, a1, false, b, (short)0, c1, false, false);
            c2 = __builtin_amdgcn_wmma_f32_16x16x4_f32(false, a2, false, b, (short)0, c2, false, false);
            c3 = __builtin_amdgcn_wmma_f32_16x16x4_f32(false, a3, false, b, (short)0, c3, false, false);
        }
        __syncthreads();
    }

    #pragma unroll
    for (int st = 0; st < 4; ++st) {
        v8f c = (st == 0) ? c0 : (st == 1) ? c1 : (st == 2) ? c2 : c3;
        #pragma unroll
        for (int r = 0; r < 8; ++r) {
            int row = m0 + (st << 4) + r + ((lane >> 4) << 3);
            if (row < M) C[(size_t)row * HID + ncol] = c[r];
        }
    }
}

// ---------- edge scatter: agg[dst] += dinv[src]*dinv[dst] * h[src]  ----------
// 16 threads per edge, 4 floats each (float4 gather, 4 float atomics).
__global__ void k_scatter(const float* __restrict__ h, float* __restrict__ agg,
                          const int* __restrict__ src, const int* __restrict__ dst,
                          const float* __restrict__ dinv, int E) {
    long long t = (long long)blockIdx.x * blockDim.x + threadIdx.x;
    long long e = t >> 4;
    if (e >= (long long)E) return;
    int f4 = (int)(t & 15) * 4;
    int s = src[e], d = dst[e];
    float norm = dinv[s] * dinv[d];
    const float4 hv = *(const float4*)(h + (size_t)s * HID + f4);
    float* ap = agg + (size_t)d * HID + f4;
    atomicAdd(ap + 0, norm * hv.x);
    atomicAdd(ap + 1, norm * hv.y);
    atomicAdd(ap + 2, norm * hv.z);
    atomicAdd(ap + 3, norm * hv.w);
}

// ---------- epilogue: agg = relu(agg + dinv[i]^2 * h + b) ----------
__global__ void k_self_bias_relu(float* __restrict__ agg, const float* __restrict__ h,
                                 const float* __restrict__ dinv, const float* __restrict__ b,
                                 int N) {
    long long idx = (long long)blockIdx.x * blockDim.x + threadIdx.x;
    if (idx >= (long long)N * HID) return;
    int i = (int)(idx >> 6);
    int f = (int)(idx & 63);
    float di = dinv[i];
    float v = agg[idx] + di * di * h[idx] + b[f];
    agg[idx] = v > 0.f ? v : 0.f;
}

// ---------- column mean over N rows -> out[64] ----------
__global__ void k_mean(const float* __restrict__ h, float* __restrict__ out, int N) {
    int tid = blockIdx.x * blockDim.x + threadIdx.x;   // 65536 threads
    int f = tid & 63;
    int slice = tid >> 6;                              // 0..1023
    float s = 0.f;
    for (int i = slice; i < N; i += 1024)
        s += h[(size_t)i * HID + f];
    atomicAdd(&out[f], s / (float)N);
}

extern "C" void kernel_launch(void* const* d_in, const int* in_sizes, int n_in,
                              void* d_out, int out_size, void* d_ws, size_t ws_size,
                              hipStream_t stream) {
    const float* x  = (const float*)d_in[0];
    const int*   ei = (const int*)d_in[1];
    const float* W1 = (const float*)d_in[2];
    const float* b1 = (const float*)d_in[3];
    const float* W2 = (const float*)d_in[4];
    const float* b2 = (const float*)d_in[5];
    float* out = (float*)d_out;

    const int IN = 2048;
    const int N  = in_sizes[0] / IN;    // 50000
    const int E  = in_sizes[1] / 2;     // 1,600,000
    const int* src = ei;
    const int* dst = ei + E;

    float* ws   = (float*)d_ws;
    const int DPAD = ((N + 63) / 64) * 64;
    float* dinv = ws;                         // [N]
    float* bufA = ws + DPAD;                  // [N x 64]
    float* bufB = bufA + (size_t)N * HID;     // [N x 64]

    const int T = 256;
    long long nh = (long long)N * HID;
    const int gemm_blocks = (N + 63) / 64;

    // --- degree / norm ---
    k_fill<<<(N + T - 1) / T, T, 0, stream>>>(dinv, N, 1.0f);           // self-loops
    k_count_deg<<<(E + T - 1) / T, T, 0, stream>>>(dst, dinv, E);
    k_rsqrt<<<(N + T - 1) / T, T, 0, stream>>>(dinv, N);

    // --- layer 1: h1 = X @ W1 ; agg = Ahat * h1 ; relu(+b1) ---
    k_gemm64<<<gemm_blocks, 128, 0, stream>>>(x, W1, bufA, N, IN);
    k_fill<<<(int)((nh + T - 1) / T), T, 0, stream>>>(bufB, nh, 0.0f);
    {
        long long threads = (long long)E * 16;
        k_scatter<<<(int)((threads + T - 1) / T), T, 0, stream>>>(bufA, bufB, src, dst, dinv, E);
    }
    k_self_bias_relu<<<(int)((nh + T - 1) / T), T, 0, stream>>>(bufB, bufA, dinv, b1, N);

    // --- layer 2: h2 = h1r @ W2 ; agg ; relu(+b2) ---
    k_gemm64<<<gemm_blocks, 128, 0, stream>>>(bufB, W2, bufA, N, HID);
    k_fill<<<(int)((nh + T - 1) / T), T, 0, stream>>>(bufB, nh, 0.0f);
    {
        long long threads = (long long)E * 16;
        k_scatter<<<(int)((threads + T - 1) / T), T, 0, stream>>>(bufA, bufB, src, dst, dinv, E);
    }
    k_self_bias_relu<<<(int)((nh + T - 1) / T), T, 0, stream>>>(bufB, bufA, dinv, b2, N);

    // --- mean over nodes ---
    k_fill<<<1, 64, 0, stream>>>(out, 64, 0.0f);
    k_mean<<<256, 256, 0, stream>>>(bufB, out, N);
}